// CRFOutputLayer_23613730193825
// MI455X (gfx1250) — compile-verified
//
#include <hip/hip_runtime.h>
#include <hip/hip_bf16.h>
#include <math.h>

typedef __attribute__((ext_vector_type(2))) float v2f;
typedef __attribute__((ext_vector_type(4))) float v4f;
typedef __attribute__((ext_vector_type(8))) float v8f;

#define B_ 64
#define T_ 512
#define D_ 1024
#define L_ 32
#define WAVES_PER_BLOCK 8

#ifndef __has_builtin
#define __has_builtin(x) 0
#endif

#if __has_builtin(__builtin_amdgcn_global_load_async_to_lds_b128)
#define USE_ASYNC_LDS 1
#else
#define USE_ASYNC_LDS 0
#endif

#if USE_ASYNC_LDS
// Param 0 of the builtin is 'int __attribute__((vector_size(16))) __device__ *'
// (probe-confirmed by the round-2 diagnostic): non-const int4* in AS(1).
typedef int v4i_gcc __attribute__((vector_size(16)));
typedef __attribute__((address_space(1))) v4i_gcc* async_gptr_t;
typedef __attribute__((address_space(3))) v4i_gcc* async_lptr_t;
#if __has_builtin(__builtin_amdgcn_s_wait_asynccnt)
#define WAIT_ASYNC(n) __builtin_amdgcn_s_wait_asynccnt(n)
#else
#define WAIT_ASYNC(n) asm volatile("s_wait_asynccnt %0" ::"i"(n))
#endif
#define COMPILER_MEM_BARRIER() asm volatile("" ::: "memory")
#endif

// ---------------------------------------------------------------------------
// Kernel 1: logits = x @ W + b  (f32 WMMA 16x16x4), fused row-softmax.
// One wave computes a 16x32 tile of logits (two 16x16 WMMA accumulators).
// x is staged into LDS with double-buffered GLOBAL_LOAD_ASYNC_TO_LDS_B128
// (ASYNCcnt-tracked) when the toolchain exposes it; otherwise via coalesced
// non-temporal B128 register staging. W (128 KB) stays hot in WGP$/L2.
// ---------------------------------------------------------------------------
__global__ __launch_bounds__(256) void logits_softmax_kernel(
    const float* __restrict__ x,      // [B*T, D]
    const float* __restrict__ W,      // [D, L]
    const float* __restrict__ bias,   // [L]
    float* __restrict__ logits_ws,    // [B*T, L]  (scratch, re-read by scan)
    float* __restrict__ probs_out)    // [B*T, L]  (output)
{
    // +4 column pad -> LDS lane stride of 36 dwords, conflict-free A reads
#if USE_ASYNC_LDS
    __shared__ float lds[WAVES_PER_BLOCK][2][16][36];   // ping-pong slabs
#else
    __shared__ float lds[WAVES_PER_BLOCK][16][36];
#endif

    const int lane = threadIdx.x & 31;
    const int wave = threadIdx.x >> 5;
    const long tile = (long)blockIdx.x * WAVES_PER_BLOCK + wave;
    const long rowbase = tile * 16;

    const int m = lane & 15;   // A-matrix row (M) for this lane
    const int h = lane >> 4;   // K half-pair selector (ISA A/B f32 layout)
    const int n = lane & 15;   // B/C column within 16-wide tile

    v8f c0 = {};  // N = 0..15
    v8f c1 = {};  // N = 16..31

#if USE_ASYNC_LDS
    // Per-lane (row, quad) assignment for the 4 async B128 copies per chunk.
    // Chunk = 16 rows x 32 cols of x = 128 float4's.
    {
        #pragma unroll
        for (int q = 0; q < 4; ++q) {
            int idx  = q * 32 + lane;
            int row  = idx >> 3;
            int quad = idx & 7;
            __builtin_amdgcn_global_load_async_to_lds_b128(
                (async_gptr_t)(x + (rowbase + row) * D_ + quad * 4),
                (async_lptr_t)&lds[wave][0][row][quad * 4], 0, 0);
        }
    }
    for (int kc = 0; kc < D_; kc += 32) {
        const int cur = (kc >> 5) & 1;
        if (kc + 32 < D_) {
            // Prefetch next chunk into the other slab (ASYNCcnt += 4)
            #pragma unroll
            for (int q = 0; q < 4; ++q) {
                int idx  = q * 32 + lane;
                int row  = idx >> 3;
                int quad = idx & 7;
                __builtin_amdgcn_global_load_async_to_lds_b128(
                    (async_gptr_t)(x + (rowbase + row) * D_ + (kc + 32) + quad * 4),
                    (async_lptr_t)&lds[wave][cur ^ 1][row][quad * 4], 0, 0);
            }
            WAIT_ASYNC(4);        // current chunk's 4 copies are complete
        } else {
            WAIT_ASYNC(0);
        }
        COMPILER_MEM_BARRIER();

        #pragma unroll
        for (int kk = 0; kk < 32; kk += 4) {
            // A 16x4 f32: lane m holds K = kk+2h, kk+2h+1 (8B aligned in LDS)
            v2f a = *(const v2f*)&lds[wave][cur][m][kk + 2 * h];
            int krow = kc + kk + 2 * h;
            // B 4x16 f32: lanes 0-15 hold K pair {k,k+1}, lanes 16-31 {k+2,k+3}
            v2f b0, b1;
            b0.x = W[(krow    ) * L_ + n];
            b0.y = W[(krow + 1) * L_ + n];
            b1.x = W[(krow    ) * L_ + 16 + n];
            b1.y = W[(krow + 1) * L_ + 16 + n];
            c0 = __builtin_amdgcn_wmma_f32_16x16x4_f32(
                     false, a, false, b0, (short)0, c0, false, false);
            c1 = __builtin_amdgcn_wmma_f32_16x16x4_f32(
                     false, a, false, b1, (short)0, c1, false, false);
        }
        COMPILER_MEM_BARRIER();
    }
#else
    for (int kc = 0; kc < D_; kc += 32) {
        // Stage a 16x32 chunk of x into LDS: 128 float4's, 4 per lane,
        // coalesced within rows; non-temporal (x is streamed exactly once).
        #pragma unroll
        for (int q = 0; q < 4; ++q) {
            int idx  = q * 32 + lane;
            int row  = idx >> 3;
            int quad = idx & 7;
            v4f val = __builtin_nontemporal_load(
                (const v4f*)(x + (rowbase + row) * D_ + kc + quad * 4));
            *(v4f*)&lds[wave][row][quad * 4] = val;
        }
        // LDS ops from the same wave are in-order: no barrier needed.
        #pragma unroll
        for (int kk = 0; kk < 32; kk += 4) {
            v2f a = *(const v2f*)&lds[wave][m][kk + 2 * h];
            int krow = kc + kk + 2 * h;
            v2f b0, b1;
            b0.x = W[(krow    ) * L_ + n];
            b0.y = W[(krow + 1) * L_ + n];
            b1.x = W[(krow    ) * L_ + 16 + n];
            b1.y = W[(krow + 1) * L_ + 16 + n];
            c0 = __builtin_amdgcn_wmma_f32_16x16x4_f32(
                     false, a, false, b0, (short)0, c0, false, false);
            c1 = __builtin_amdgcn_wmma_f32_16x16x4_f32(
                     false, a, false, b1, (short)0, c1, false, false);
        }
    }
#endif

    // Bias epilogue
    const float bv0 = bias[n];
    const float bv1 = bias[16 + n];
    #pragma unroll
    for (int r = 0; r < 8; ++r) { c0[r] += bv0; c1[r] += bv1; }

    // C layout: VGPR r -> row M=r (lanes 0-15) / M=8+r (lanes 16-31), N=lane%16.
    // shfl_xor with masks < 16 reduces exactly within one matrix row.
    #pragma unroll
    for (int r = 0; r < 8; ++r) {
        long row = rowbase + h * 8 + r;
        float a0 = c0[r], a1 = c1[r];
        logits_ws[row * L_ + n]      = a0;   // regular store: re-read soon (L2)
        logits_ws[row * L_ + 16 + n] = a1;

        float rm = fmaxf(a0, a1);
        #pragma unroll
        for (int off = 1; off < 16; off <<= 1)
            rm = fmaxf(rm, __shfl_xor(rm, off, 32));
        float e0 = expf(a0 - rm);
        float e1 = expf(a1 - rm);
        float s = e0 + e1;
        #pragma unroll
        for (int off = 1; off < 16; off <<= 1)
            s += __shfl_xor(s, off, 32);
        float inv = 1.0f / s;
        __builtin_nontemporal_store(e0 * inv, &probs_out[row * L_ + n]);
        __builtin_nontemporal_store(e1 * inv, &probs_out[row * L_ + 16 + n]);
    }
}

// ---------------------------------------------------------------------------
// Kernel 2: CRF forward scan + gold score. One wave per batch item; lane = tag.
// logsumexp_i(alpha_i + T_ij) == m + log(sum_i exp(alpha_i - m) * E_ij),
// with E = exp(transitions) held in registers -> one exp + 32 FMA per step.
// ---------------------------------------------------------------------------
__global__ __launch_bounds__(32) void crf_scan_kernel(
    const float* __restrict__ logits,        // [B, T, L] from ws
    const unsigned char* __restrict__ mask,  // [B, T]
    const int* __restrict__ labels,          // [B, T]
    const float* __restrict__ trans,         // [L, L]
    const float* __restrict__ start_t,       // [L]
    const float* __restrict__ end_t,         // [L]
    float* __restrict__ partial)             // [B] gold - logZ
{
    const int b = blockIdx.x;
    const int j = threadIdx.x;               // this lane's tag index

    // Column j of exp(transitions): E[i][j] for i = 0..31
    float Ecol[32];
    #pragma unroll
    for (int i = 0; i < 32; ++i) Ecol[i] = expf(trans[i * L_ + j]);

    const float* lg = logits + (long)b * T_ * L_;
    const unsigned char* mk = mask + (long)b * T_;
    const int* tg = labels + (long)b * T_;

    float logit = lg[j];                     // t = 0 emission vector
    float alpha = start_t[j] + logit;

    int   ptag = tg[0];
    float pmf  = mk[0] ? 1.0f : 0.0f;
    float gold = start_t[ptag];              // uniform across lanes

    for (int t = 1; t < T_; ++t) {
        // gold: emission at t-1 (covers t-1 = 0..T-2) and transition into t
        gold += __shfl(logit, ptag, 32) * pmf;
        float nl  = lg[t * L_ + j];
        int   tag = tg[t];
        float mf  = mk[t] ? 1.0f : 0.0f;
        gold += trans[ptag * L_ + tag] * mf;

        // alpha_j' = m + log( sum_i exp(alpha_i - m) * E_ij ) + logit_j
        float m = alpha;
        #pragma unroll
        for (int off = 1; off < 32; off <<= 1)
            m = fmaxf(m, __shfl_xor(m, off, 32));
        float p = expf(alpha - m);
        float dot = 0.0f;
        #pragma unroll
        for (int i = 0; i < 32; ++i)
            dot = fmaf(__shfl(p, i, 32), Ecol[i], dot);
        float na = m + logf(dot) + nl;
        alpha = (mf != 0.0f) ? na : alpha;

        ptag = tag; pmf = mf; logit = nl;
    }

    // lengths = sum(mask); last tag + end/last-emission terms
    int cnt = 0;
    for (int t = j; t < T_; t += 32) cnt += mk[t] ? 1 : 0;
    #pragma unroll
    for (int off = 1; off < 32; off <<= 1) cnt += __shfl_xor(cnt, off, 32);
    int last = tg[cnt - 1];
    gold += end_t[last];
    gold += __shfl(logit, last, 32) * (mk[T_ - 1] ? 1.0f : 0.0f);

    // logZ = logsumexp(alpha + end_t)
    float v = alpha + end_t[j];
    float m2 = v;
    #pragma unroll
    for (int off = 1; off < 32; off <<= 1)
        m2 = fmaxf(m2, __shfl_xor(m2, off, 32));
    float s2 = expf(v - m2);
    #pragma unroll
    for (int off = 1; off < 32; off <<= 1)
        s2 += __shfl_xor(s2, off, 32);
    float logZ = m2 + logf(s2);

    if (j == 0) partial[b] = gold - logZ;
}

// ---------------------------------------------------------------------------
// Kernel 3: loss = -sum_b (gold_b - logZ_b), deterministic single-wave reduce.
// ---------------------------------------------------------------------------
__global__ __launch_bounds__(32) void finalize_kernel(
    const float* __restrict__ partial, float* __restrict__ loss_out)
{
    int j = threadIdx.x;
    float v = partial[j] + partial[j + 32];
    #pragma unroll
    for (int off = 1; off < 32; off <<= 1) v += __shfl_xor(v, off, 32);
    if (j == 0) *loss_out = -v;
}

extern "C" void kernel_launch(void* const* d_in, const int* in_sizes, int n_in,
                              void* d_out, int out_size, void* d_ws, size_t ws_size,
                              hipStream_t stream) {
    const float*         x       = (const float*)d_in[0];
    const unsigned char* mask    = (const unsigned char*)d_in[1]; // bool
    const int*           labels  = (const int*)d_in[2];
    const float*         W       = (const float*)d_in[3];
    const float*         bias    = (const float*)d_in[4];
    const float*         trans   = (const float*)d_in[5];
    const float*         start_t = (const float*)d_in[6];
    const float*         end_t   = (const float*)d_in[7];

    float* probs = (float*)d_out;                          // [B*T*L]
    float* loss  = probs + (size_t)B_ * T_ * L_;           // [1]

    float* logits_ws = (float*)d_ws;                       // [B*T*L]
    float* partial   = logits_ws + (size_t)B_ * T_ * L_;   // [B]

    const int tiles  = (B_ * T_) / 16;                     // 2048
    const int blocks = tiles / WAVES_PER_BLOCK;            // 256

    logits_softmax_kernel<<<dim3(blocks), dim3(256), 0, stream>>>(
        x, W, bias, logits_ws, probs);
    crf_scan_kernel<<<dim3(B_), dim3(32), 0, stream>>>(
        logits_ws, mask, labels, trans, start_t, end_t, partial);
    finalize_kernel<<<dim3(1), dim3(32), 0, stream>>>(partial, loss);
}